// GraphEncoder4Link_68771016343679
// MI455X (gfx1250) — compile-verified
//
#include <hip/hip_runtime.h>
#include <hip/hip_bf16.h>

// GCN encoder for MI455X (gfx1250, wave32).
// Dense GEMMs -> V_WMMA_F32_16X16X4_F32 (exact fp32, matches reference math).
// Sparse aggregation -> global_atomic_add_f32 into an L2-resident buffer
// (25.6 MB per feature matrix << 192 MB L2), self-loop term written
// non-atomically first so no extra zeroing pass is needed.

#define NEG_SLOPE 0.01f

typedef __attribute__((ext_vector_type(2))) float v2f;
typedef __attribute__((ext_vector_type(8))) float v8f;

__device__ __forceinline__ float lrelu_f(float v) {
  return v > 0.0f ? v : v * NEG_SLOPE;
}

// ---------------- degree / normalization ----------------
__global__ void k_fill1(float* __restrict__ deg, int N) {
  int i = blockIdx.x * blockDim.x + threadIdx.x;
  if (i < N) deg[i] = 1.0f;  // self-loop contribution to in-degree
}

__global__ void k_count(const long long* __restrict__ dst, float* __restrict__ deg, int E) {
  int e = blockIdx.x * blockDim.x + threadIdx.x;
  if (e < E) atomicAdd(deg + (int)dst[e], 1.0f);
}

__global__ void k_rsqrt_ip(float* __restrict__ deg, int N) {
  int i = blockIdx.x * blockDim.x + threadIdx.x;
  if (i < N) deg[i] = rsqrtf(deg[i]);  // deg >= 1 always (self loops)
}

// ---------------- WMMA fp32 GEMM: Y[N,64] = X[N,64] @ W[64,64] (+bias)(+lrelu) ----
// Block = 128 threads = 4 waves; each wave owns one 16-row strip and computes
// a 16x64 output via 4 accumulator tiles x 16 K-steps of V_WMMA_F32_16X16X4_F32.
template <bool BIAS, bool LRELU>
__global__ void k_gemm64(const float* __restrict__ X, const float* __restrict__ W,
                         const float* __restrict__ bias, float* __restrict__ Y, int N) {
  __shared__ float sW[64 * 64];
  int t = threadIdx.x;
  for (int i = t; i < 64 * 64; i += 128) sW[i] = W[i];
  __syncthreads();

  const int wave = t >> 5;
  const int lane = t & 31;
  const int strip = blockIdx.x * 4 + wave;   // 16-row strip index
  const int row0 = strip * 16;
  if (row0 >= N) return;                      // uniform per wave

  const int lrow  = lane & 15;                // M (for A) / N (for B) within tile
  const int khalf = lane >> 4;                // 0: K={kc,kc+1}, 1: K={kc+2,kc+3}
  int arow = row0 + lrow;
  if (arow >= N) arow = N - 1;                // clamp loads for partial strips

  v8f acc0 = {}, acc1 = {}, acc2 = {}, acc3 = {};

#pragma unroll
  for (int kc = 0; kc < 64; kc += 4) {
    const int kk = kc + khalf * 2;
    // A 16x4 fragment (ISA 7.12.2: lanes 0-15 K=kc..kc+1, lanes 16-31 K=kc+2..kc+3)
    v2f a;
    a.x = X[(long long)arow * 64 + kk];
    a.y = X[(long long)arow * 64 + kk + 1];
    // B 4x16 fragments, one per 16-col tile (mirrored layout, rows striped across lanes)
    v2f b0, b1, b2, b3;
    b0.x = sW[kk * 64 +  0 + lrow]; b0.y = sW[(kk + 1) * 64 +  0 + lrow];
    b1.x = sW[kk * 64 + 16 + lrow]; b1.y = sW[(kk + 1) * 64 + 16 + lrow];
    b2.x = sW[kk * 64 + 32 + lrow]; b2.y = sW[(kk + 1) * 64 + 32 + lrow];
    b3.x = sW[kk * 64 + 48 + lrow]; b3.y = sW[(kk + 1) * 64 + 48 + lrow];

    acc0 = __builtin_amdgcn_wmma_f32_16x16x4_f32(false, a, false, b0, (short)0, acc0, false, false);
    acc1 = __builtin_amdgcn_wmma_f32_16x16x4_f32(false, a, false, b1, (short)0, acc1, false, false);
    acc2 = __builtin_amdgcn_wmma_f32_16x16x4_f32(false, a, false, b2, (short)0, acc2, false, false);
    acc3 = __builtin_amdgcn_wmma_f32_16x16x4_f32(false, a, false, b3, (short)0, acc3, false, false);
  }

  // C/D layout: VGPR r -> (M = r, N = lane) for lanes 0-15, (M = r+8, N = lane-16) for 16-31
  v8f accs[4] = {acc0, acc1, acc2, acc3};
#pragma unroll
  for (int nt = 0; nt < 4; ++nt) {
#pragma unroll
    for (int r = 0; r < 8; ++r) {
      const int orow = row0 + r + khalf * 8;
      if (orow < N) {
        const int ocol = nt * 16 + lrow;
        float v = accs[nt][r];
        if (BIAS) v += bias[ocol];
        if (LRELU) v = lrelu_f(v);
        Y[(long long)orow * 64 + ocol] = v;
      }
    }
  }
}

// ---------------- aggregation ----------------
// out[i,:] = hw[i,:] * dinv[i]^2   (self-loop term, also initializes out)
__global__ void k_selfinit(const float* __restrict__ hw, const float* __restrict__ dinv,
                           float* __restrict__ out, int N) {
  long long gid = (long long)blockIdx.x * blockDim.x + threadIdx.x;
  int i = (int)(gid >> 4), q = (int)(gid & 15);
  if (i >= N) return;
  float di = dinv[i];
  float s = di * di;
  float4 v = ((const float4*)(hw + (long long)i * 64))[q];
  float4 o = make_float4(v.x * s, v.y * s, v.z * s, v.w * s);
  ((float4*)(out + (long long)i * 64))[q] = o;
}

// out[dst,:] += hw[src,:] * dinv[src]*dinv[dst]  -- 16 lanes per edge, float4 each
__global__ void k_edge(const long long* __restrict__ src, const long long* __restrict__ dst,
                       const float* __restrict__ dinv, const float* __restrict__ hw,
                       float* __restrict__ out, int E) {
  long long gid = (long long)blockIdx.x * blockDim.x + threadIdx.x;
  int e = (int)(gid >> 4), q = (int)(gid & 15);
  if (e >= E) return;
  int s = (int)src[e];
  int d = (int)dst[e];
  float nrm = dinv[s] * dinv[d];
  float4 v = ((const float4*)(hw + (long long)s * 64))[q];
  float* o = out + (long long)d * 64 + q * 4;
  atomicAdd(o + 0, v.x * nrm);
  atomicAdd(o + 1, v.y * nrm);
  atomicAdd(o + 2, v.z * nrm);
  atomicAdd(o + 3, v.w * nrm);
}

// y = lrelu(y + bias + (res ? res : 0))   (in place)
__global__ void k_finish(float* __restrict__ y, const float* __restrict__ bias,
                         const float* __restrict__ res, int N) {
  long long gid = (long long)blockIdx.x * blockDim.x + threadIdx.x;
  int i = (int)(gid >> 4), q = (int)(gid & 15);
  if (i >= N) return;
  float4 b = ((const float4*)bias)[q];
  float4 v = ((float4*)(y + (long long)i * 64))[q];
  v.x += b.x; v.y += b.y; v.z += b.z; v.w += b.w;
  if (res) {
    float4 r = ((const float4*)(res + (long long)i * 64))[q];
    v.x += r.x; v.y += r.y; v.z += r.z; v.w += r.w;
  }
  v.x = lrelu_f(v.x); v.y = lrelu_f(v.y); v.z = lrelu_f(v.z); v.w = lrelu_f(v.w);
  ((float4*)(y + (long long)i * 64))[q] = v;
}

// y = lrelu(a + y)
__global__ void k_addlrelu(const float* __restrict__ a, float* __restrict__ y, long long n4) {
  long long gid = (long long)blockIdx.x * blockDim.x + threadIdx.x;
  if (gid >= n4) return;
  float4 av = ((const float4*)a)[gid];
  float4 yv = ((float4*)y)[gid];
  yv.x = lrelu_f(av.x + yv.x); yv.y = lrelu_f(av.y + yv.y);
  yv.z = lrelu_f(av.z + yv.z); yv.w = lrelu_f(av.w + yv.w);
  ((float4*)y)[gid] = yv;
}

extern "C" void kernel_launch(void* const* d_in, const int* in_sizes, int n_in,
                              void* d_out, int out_size, void* d_ws, size_t ws_size,
                              hipStream_t stream) {
  const float*      x   = (const float*)d_in[0];
  const long long*  ei  = (const long long*)d_in[1];
  const float*      W1  = (const float*)d_in[2];
  const float*      b1  = (const float*)d_in[3];
  const float*      W2  = (const float*)d_in[4];
  const float*      b2  = (const float*)d_in[5];
  const float*      Wm1 = (const float*)d_in[6];
  const float*      bm1 = (const float*)d_in[7];
  const float*      Wm2 = (const float*)d_in[8];
  const float*      bm2 = (const float*)d_in[9];

  const int N = in_sizes[0] / 64;
  const int E = in_sizes[1] / 2;
  const long long* src = ei;
  const long long* dst = ei + E;

  float* out = (float*)d_out;             // also serves as residual buffer "B"
  float* dinv = (float*)d_ws;             // N floats (deg -> dinv in place)
  float* hw   = dinv + N;                 // N*64 (GEMM result / scratch)
  float* A    = hw + (size_t)N * 64;      // N*64 (h after conv1 / m2)

  const int TB = 256;
  const int gN   = (N + TB - 1) / TB;
  const int gE   = (E + TB - 1) / TB;
  const int gN16 = (int)(((long long)N * 16 + TB - 1) / TB);
  const int gE16 = (int)(((long long)E * 16 + TB - 1) / TB);
  const int strips = (N + 15) / 16;
  const int gGemm  = (strips + 3) / 4;

  // normalization: deg -> dinv
  k_fill1<<<gN, TB, 0, stream>>>(dinv, N);
  k_count<<<gE, TB, 0, stream>>>(dst, dinv, E);
  k_rsqrt_ip<<<gN, TB, 0, stream>>>(dinv, N);

  // conv1: A = lrelu(aggregate(x @ W1) + b1)
  k_gemm64<false, false><<<gGemm, 128, 0, stream>>>(x, W1, nullptr, hw, N);
  k_selfinit<<<gN16, TB, 0, stream>>>(hw, dinv, A, N);
  k_edge<<<gE16, TB, 0, stream>>>(src, dst, dinv, hw, A, E);
  k_finish<<<gN16, TB, 0, stream>>>(A, b1, nullptr, N);

  // conv2: out = lrelu(aggregate(A @ W2) + b2 + A)
  k_gemm64<false, false><<<gGemm, 128, 0, stream>>>(A, W2, nullptr, hw, N);
  k_selfinit<<<gN16, TB, 0, stream>>>(hw, dinv, out, N);
  k_edge<<<gE16, TB, 0, stream>>>(src, dst, dinv, hw, out, E);
  k_finish<<<gN16, TB, 0, stream>>>(out, b2, A, N);

  // MLP: hw = lrelu(out @ Wm1 + bm1); A = hw @ Wm2 + bm2; out = lrelu(A + out)
  k_gemm64<true, true ><<<gGemm, 128, 0, stream>>>(out, Wm1, bm1, hw, N);
  k_gemm64<true, false><<<gGemm, 128, 0, stream>>>(hw, Wm2, bm2, A, N);
  k_addlrelu<<<gN16, TB, 0, stream>>>(A, out, (long long)N * 16);
}